// action_network_51548197487385
// MI455X (gfx1250) — compile-verified
//
#include <hip/hip_runtime.h>

#define N_NODES 50000
#define N_EDGES 25000
#define NNZ     600000
#define DFEAT   128
#define HFEAT   256
#define DEPTH   2

typedef float v2f __attribute__((ext_vector_type(2)));
typedef float v8f __attribute__((ext_vector_type(8)));

// ---------------------------------------------------------------------------
// Degree / reciprocal helpers
// ---------------------------------------------------------------------------
__global__ void __launch_bounds__(256) degree_kernel(const int* __restrict__ idx,
                                                     float* __restrict__ cnt, int nnz) {
    int i = blockIdx.x * blockDim.x + threadIdx.x;
    if (i < nnz) atomicAdd(&cnt[idx[i]], 1.0f);
}

__global__ void __launch_bounds__(256) recip_kernel(float* __restrict__ cnt, int n) {
    int i = blockIdx.x * blockDim.x + threadIdx.x;
    if (i < n) cnt[i] = 1.0f / fmaxf(cnt[i], 1.0f);
}

// ---------------------------------------------------------------------------
// Fused GEMM: C[N,MOUT] = relu(A[N,K] @ W[MOUT,K]^T + bias [+ skip*rdeg[row]])
// Full-precision f32 WMMA (V_WMMA_F32_16X16X4_F32), 16-row tile per block,
// A tile staged in LDS (padded row stride K+4 -> conflict-free ds_load_b64),
// 8 waves per block, each wave owns 16-column tiles, 2 accumulator chains.
// ---------------------------------------------------------------------------
template <int K, int MOUT, bool FUSE_SKIP>
__global__ void __launch_bounds__(256)
gemm_relu_kernel(const float* __restrict__ A, const float* __restrict__ W,
                 const float* __restrict__ bias,
                 const float* __restrict__ skip, const float* __restrict__ rdeg,
                 float* __restrict__ C) {
    constexpr int LDA = K + 4;                 // padded LDS stride (dwords), 16B aligned
    __shared__ float As[16 * LDA];

    const int rowBase = blockIdx.x * 16;
    const int tid     = threadIdx.x;

    // Cooperative 16xK tile load (float4 granularity)
    constexpr int NV4 = 16 * K / 4;
    for (int i = tid; i < NV4; i += 256) {
        const int row = i / (K / 4);
        const int c4  = i % (K / 4);
        const float4 v = *(const float4*)(A + (size_t)(rowBase + row) * K + 4 * c4);
        *(float4*)&As[row * LDA + 4 * c4] = v;
    }
    __syncthreads();

    const int lane = tid & 31;
    const int wave = tid >> 5;      // 0..7
    const int half = lane >> 4;     // 0: lanes 0-15, 1: lanes 16-31
    const int lr   = lane & 15;

    const float* arow = As + lr * LDA;

    for (int ct = wave; ct < MOUT / 16; ct += 8) {
        const int colBase = ct * 16;
        const float* wrow = W + (size_t)(colBase + lr) * K;

        v8f c0 = {}, c1 = {};
        #pragma unroll 4
        for (int k0 = 0; k0 < K; k0 += 8) {
            // A frag (16x4 f32): lanes 0-15 hold K = k0,k0+1 ; lanes 16-31 K = k0+2,k0+3
            v2f a0 = *(const v2f*)&arow[k0 + 2 * half];
            v2f b0 = *(const v2f*)&wrow[k0 + 2 * half];
            v2f a1 = *(const v2f*)&arow[k0 + 4 + 2 * half];
            v2f b1 = *(const v2f*)&wrow[k0 + 4 + 2 * half];
            c0 = __builtin_amdgcn_wmma_f32_16x16x4_f32(false, a0, false, b0,
                                                       (short)0, c0, false, false);
            c1 = __builtin_amdgcn_wmma_f32_16x16x4_f32(false, a1, false, b1,
                                                       (short)0, c1, false, false);
        }

        const float bcol = bias[colBase + lr];
        #pragma unroll
        for (int r = 0; r < 8; ++r) {
            const int row = rowBase + r + 8 * half;   // C layout: VGPR r -> M=r / M=r+8
            const int col = colBase + lr;
            float v = c0[r] + c1[r] + bcol;
            if constexpr (FUSE_SKIP)
                v += skip[(size_t)row * MOUT + col] * rdeg[row];
            C[(size_t)row * MOUT + col] = fmaxf(v, 0.0f);
        }
    }
}

// ---------------------------------------------------------------------------
// Wave-per-incidence scatter:  dst[sidx[i]] += src[gidx[i]] * (gscale?gscale[gidx[i]]:1)
// 32 lanes x float4 = one 128-float feature row per wave.
// ---------------------------------------------------------------------------
template <bool SCALE>
__global__ void __launch_bounds__(256)
scatter_add_kernel(const float* __restrict__ src, const int* __restrict__ gidx,
                   const int* __restrict__ sidx, const float* __restrict__ gscale,
                   float* __restrict__ dst, int nnz) {
    const int lane   = threadIdx.x & 31;
    const int wave   = (blockIdx.x * blockDim.x + threadIdx.x) >> 5;
    const int nwaves = (gridDim.x * blockDim.x) >> 5;
    for (int i = wave; i < nnz; i += nwaves) {
        const int g = gidx[i];
        const int s = sidx[i];
        float4 v = ((const float4*)(src + (size_t)g * DFEAT))[lane];
        if constexpr (SCALE) {
            const float sc = gscale[g];
            v.x *= sc; v.y *= sc; v.z *= sc; v.w *= sc;
        }
        float* d = dst + (size_t)s * DFEAT + lane * 4;
        atomicAdd(d + 0, v.x);
        atomicAdd(d + 1, v.y);
        atomicAdd(d + 2, v.z);
        atomicAdd(d + 3, v.w);
    }
}

// ---------------------------------------------------------------------------
// Head: a = h @ Wa^T + ba ; log_softmax over 4 classes. One wave per node.
// ---------------------------------------------------------------------------
__global__ void __launch_bounds__(256)
head_kernel(const float* __restrict__ h, const float* __restrict__ Wa,
            const float* __restrict__ ba, float* __restrict__ out, int n) {
    const int lane   = threadIdx.x & 31;
    const int wave   = (blockIdx.x * blockDim.x + threadIdx.x) >> 5;
    const int nwaves = (gridDim.x * blockDim.x) >> 5;
    for (int node = wave; node < n; node += nwaves) {
        const float4 xv = ((const float4*)(h + (size_t)node * DFEAT))[lane];
        float acc[4];
        #pragma unroll
        for (int j = 0; j < 4; ++j) {
            const float4 wv = ((const float4*)(Wa + (size_t)j * DFEAT))[lane];
            acc[j] = xv.x * wv.x + xv.y * wv.y + xv.z * wv.z + xv.w * wv.w;
        }
        #pragma unroll
        for (int off = 16; off > 0; off >>= 1) {
            #pragma unroll
            for (int j = 0; j < 4; ++j) acc[j] += __shfl_down(acc[j], off, 32);
        }
        if (lane == 0) {
            float a0 = acc[0] + ba[0], a1 = acc[1] + ba[1];
            float a2 = acc[2] + ba[2], a3 = acc[3] + ba[3];
            const float mx  = fmaxf(fmaxf(a0, a1), fmaxf(a2, a3));
            const float lse = mx + logf(expf(a0 - mx) + expf(a1 - mx) +
                                        expf(a2 - mx) + expf(a3 - mx));
            float* o = out + (size_t)node * 4;
            o[0] = a0 - lse; o[1] = a1 - lse; o[2] = a2 - lse; o[3] = a3 - lse;
        }
    }
}

// ---------------------------------------------------------------------------
extern "C" void kernel_launch(void* const* d_in, const int* in_sizes, int n_in,
                              void* d_out, int out_size, void* d_ws, size_t ws_size,
                              hipStream_t stream) {
    const float* x     = (const float*)d_in[0];
    const int*   v_idx = (const int*)  d_in[1];
    const int*   e_idx = (const int*)  d_in[2];
    const float* W1    = (const float*)d_in[3];
    const float* b1    = (const float*)d_in[4];
    const float* W2    = (const float*)d_in[5];
    const float* b2    = (const float*)d_in[6];
    const float* Wu    = (const float*)d_in[7];
    const float* bu    = (const float*)d_in[8];
    const float* Wa    = (const float*)d_in[9];
    const float* ba    = (const float*)d_in[10];
    float* out = (float*)d_out;

    char*  ws  = (char*)d_ws;
    size_t off = 0;
    auto alloc = [&](size_t bytes) -> void* {
        void* p = ws + off;
        off += (bytes + 255) & ~(size_t)255;
        return p;
    };
    float* hbuf  = (float*)alloc((size_t)N_NODES * DFEAT * 4);
    float* m1    = (float*)alloc((size_t)N_NODES * HFEAT * 4);
    float* m2    = (float*)alloc((size_t)N_NODES * DFEAT * 4);
    float* e_acc = (float*)alloc((size_t)N_EDGES * DFEAT * 4);
    float* v_acc = (float*)alloc((size_t)N_NODES * DFEAT * 4);
    float* redeg = (float*)alloc((size_t)N_EDGES * 4);
    float* rvdeg = (float*)alloc((size_t)N_NODES * 4);

    // h = x (inputs must not be mutated; update GEMM runs in-place on hbuf)
    hipMemcpyAsync(hbuf, x, (size_t)N_NODES * DFEAT * 4, hipMemcpyDeviceToDevice, stream);

    // Degrees -> reciprocals of max(deg,1); identical every depth, compute once.
    hipMemsetAsync(redeg, 0, (size_t)N_EDGES * 4, stream);
    hipMemsetAsync(rvdeg, 0, (size_t)N_NODES * 4, stream);
    degree_kernel<<<(NNZ + 255) / 256, 256, 0, stream>>>(e_idx, redeg, NNZ);
    degree_kernel<<<(NNZ + 255) / 256, 256, 0, stream>>>(v_idx, rvdeg, NNZ);
    recip_kernel<<<(N_EDGES + 255) / 256, 256, 0, stream>>>(redeg, N_EDGES);
    recip_kernel<<<(N_NODES + 255) / 256, 256, 0, stream>>>(rvdeg, N_NODES);

    const int rowBlocks = N_NODES / 16;   // 50000 = 16 * 3125, exact
    for (int depth = 0; depth < DEPTH; ++depth) {
        // m1 = relu(h @ W1^T + b1)          [N,128]x[128,256]
        gemm_relu_kernel<DFEAT, HFEAT, false>
            <<<rowBlocks, 256, 0, stream>>>(hbuf, W1, b1, nullptr, nullptr, m1);
        // m2 = relu(m1 @ W2^T + b2)         [N,256]x[256,128]
        gemm_relu_kernel<HFEAT, DFEAT, false>
            <<<rowBlocks, 256, 0, stream>>>(m1, W2, b2, nullptr, nullptr, m2);

        // e_acc[e] = sum_{i: e_idx[i]=e} m2[v_idx[i]]
        hipMemsetAsync(e_acc, 0, (size_t)N_EDGES * DFEAT * 4, stream);
        scatter_add_kernel<false>
            <<<2048, 256, 0, stream>>>(m2, v_idx, e_idx, nullptr, e_acc, NNZ);
        // v_acc[v] = sum_{i: v_idx[i]=v} e_acc[e_idx[i]] * redeg[e_idx[i]]
        hipMemsetAsync(v_acc, 0, (size_t)N_NODES * DFEAT * 4, stream);
        scatter_add_kernel<true>
            <<<2048, 256, 0, stream>>>(e_acc, e_idx, v_idx, redeg, v_acc, NNZ);

        // h = relu(h @ Wu^T + bu + v_acc * rvdeg[row])   (safe in-place)
        gemm_relu_kernel<DFEAT, DFEAT, true>
            <<<rowBlocks, 256, 0, stream>>>(hbuf, Wu, bu, v_acc, rvdeg, hbuf);
    }

    head_kernel<<<1024, 256, 0, stream>>>(hbuf, Wa, ba, out, N_NODES);
}